// Patcher_76484777607757
// MI455X (gfx1250) — compile-verified
//
#include <hip/hip_runtime.h>
#include <hip/hip_bf16.h>

// Problem constants (match reference setup_inputs / module constants)
#define BATCH 16
#define NBOX  16
#define IMH   640
#define IMW   640
#define PH    300
#define PW    300
#define AREA_THRESH 900

// 8-byte float vector with 4-byte alignment (pixel stride is 12B -> only 4B aligned)
typedef float v2f __attribute__((ext_vector_type(2))) __attribute__((aligned(4)));

// Workspace record per (b,n): {yp_i, xp_i, ph_i, pw_i, valid, 0, 0, 0} -> 32 bytes
// Per image: NBOX * 32 = 512 bytes (exactly 32 x b128 async transfers).

__global__ __launch_bounds__(256)
void patch_setup_kernel(const float* __restrict__ boxes,   // [B,N,4] (ymin,xmin,ymax,xmax)
                        const int*   __restrict__ dec,     // [B,N,3]
                        float* __restrict__ out_pb,        // [B,N,4] (yp,xp,ph,pw)
                        float* __restrict__ out_dec,       // [B,N,3] as float
                        int*   __restrict__ ws_box)        // [B,N,8]
{
    // XLA rounds every HLO op separately; FMA contraction of e.g. 640 - 0.2*h
    // changes the value by 1 ULP and flips (int) casts / threshold decisions.
    #pragma clang fp contract(off)

    const int t = threadIdx.x;              // 0..255 == b*NBOX + n

    const float ymin = boxes[t * 4 + 0];
    const float xmin = boxes[t * 4 + 1];
    const float ymax = boxes[t * 4 + 2];
    const float xmax = boxes[t * 4 + 3];

    const float h  = ymax - ymin;
    const float w  = xmax - xmin;
    const float ph = h * 0.2f;              // SCALE
    const float pw = ph;                    // ASPECT(=1.0) * ph == ph exactly
    float yp = ymin + h * 0.5f;             // ORIGIN
    float xp = xmin + w * 0.5f;
    if (yp + ph > (float)IMH) yp = (float)IMH - ph;
    if (xp + pw > (float)IMW) xp = (float)IMW - pw;

    out_pb[t * 4 + 0] = yp;
    out_pb[t * 4 + 1] = xp;
    out_pb[t * 4 + 2] = ph;
    out_pb[t * 4 + 3] = pw;

    const int area  = (int)(ph * pw);       // astype(int32): truncate (>= 0)
    const int valid = (area > AREA_THRESH) ? 1 : 0;

    int* r = ws_box + t * 8;
    r[0] = (int)yp; r[1] = (int)xp; r[2] = (int)ph; r[3] = (int)pw;
    r[4] = valid;   r[5] = 0;       r[6] = 0;       r[7] = 0;

    // transform_decisions passed through (cast to output float dtype)
    for (int i = t; i < BATCH * NBOX * 3; i += 256)
        out_dec[i] = (float)dec[i];
}

__global__ __launch_bounds__(256)
void patch_paste_kernel(const float* __restrict__ images,  // [B,H,W,3]
                        const float* __restrict__ patch,   // [PH,PW,3]
                        const int*   __restrict__ ws_box,  // [B,N,8]
                        float* __restrict__ out)           // [B,H,W,3]
{
    // sy/sx feed floor() -> texel select; keep per-op RNE rounding like XLA.
    #pragma clang fp contract(off)

    __shared__ int sbox[NBOX * 8];          // 512 bytes of box metadata for this image

    const int blocks_per_img = (IMH * IMW) / 256;          // 1600
    const int b   = blockIdx.x / blocks_per_img;
    const int pix = (blockIdx.x % blocks_per_img) * 256 + threadIdx.x;

    // ---- CDNA5 async global->LDS staging of this image's 16 box records ----
    // 32 lanes x b128 = 512 bytes, tracked on ASYNCcnt (CDNA5 ISA sec.4).
    if (threadIdx.x < 32) {
        const char* g = (const char*)(ws_box + b * NBOX * 8) + threadIdx.x * 16;
        __attribute__((address_space(3))) char* l =
            (__attribute__((address_space(3))) char*)sbox + threadIdx.x * 16;
        asm volatile("global_load_async_to_lds_b128 %0, %1, off"
                     :: "v"(l), "v"(g) : "memory");
    }
    if (threadIdx.x == 0) {
        // warm L2/WGP$ for the (cache-resident, heavily reused) patch texels
        __builtin_prefetch(patch, 0, 1);
    }
    asm volatile("s_wait_asynccnt 0" ::: "memory");        // issuing wave drains its copies
    __syncthreads();                                       // publish LDS to all waves

    const int y = pix / IMW;
    const int x = pix % IMW;

    const size_t base = (((size_t)b * IMH + y) * IMW + x) * 3;

    // Streaming (read-once) image data: non-temporal loads keep the reused
    // patch in WGP$/L2 instead of 78.6MB of single-use lines.
    v2f   in01 = __builtin_nontemporal_load((const v2f*)(images + base));
    float c2   = __builtin_nontemporal_load(images + base + 2);
    float c0 = in01.x;
    float c1 = in01.y;

    // Last-write-wins == highest-index valid covering box (the resized value is
    // independent of image content, so the sequential scan collapses to this).
    for (int n = NBOX - 1; n >= 0; --n) {
        const int yp = sbox[n * 8 + 0];
        const int xp = sbox[n * 8 + 1];
        const int ph = sbox[n * 8 + 2];
        const int pw = sbox[n * 8 + 3];
        const int v  = sbox[n * 8 + 4];
        if (v && y >= yp && y < yp + ph && x >= xp && x < xp + pw) {
            const float phf = (float)(ph > 1 ? ph : 1);
            const float pwf = (float)(pw > 1 ? pw : 1);
            float sy = ((float)(y - yp) + 0.5f) * ((float)PH / phf) - 0.5f;
            float sx = ((float)(x - xp) + 0.5f) * ((float)PW / pwf) - 0.5f;
            sy = fminf(fmaxf(sy, 0.0f), (float)(PH - 1));
            sx = fminf(fmaxf(sx, 0.0f), (float)(PW - 1));
            const int y0 = (int)sy;                        // sy >= 0 -> trunc == floor
            const int x0 = (int)sx;
            const int y1 = (y0 + 1 < PH) ? y0 + 1 : PH - 1;
            const int x1 = (x0 + 1 < PW) ? x0 + 1 : PW - 1;
            const float wy = sy - (float)y0;
            const float wx = sx - (float)x0;
            const float w00 = (1.0f - wy) * (1.0f - wx);
            const float w01 = (1.0f - wy) * wx;
            const float w10 = wy * (1.0f - wx);
            const float w11 = wy * wx;
            const float* p00 = patch + ((size_t)y0 * PW + x0) * 3;
            const float* p01 = patch + ((size_t)y0 * PW + x1) * 3;
            const float* p10 = patch + ((size_t)y1 * PW + x0) * 3;
            const float* p11 = patch + ((size_t)y1 * PW + x1) * 3;
            c0 = w00 * p00[0] + w01 * p01[0] + w10 * p10[0] + w11 * p11[0];
            c1 = w00 * p00[1] + w01 * p01[1] + w10 * p10[1] + w11 * p11[1];
            c2 = w00 * p00[2] + w01 * p01[2] + w10 * p10[2] + w11 * p11[2];
            break;
        }
    }

    // Write-once output: non-temporal stores (th:*_NT far policy).
    v2f out01; out01.x = c0; out01.y = c1;
    __builtin_nontemporal_store(out01, (v2f*)(out + base));
    __builtin_nontemporal_store(c2, out + base + 2);
}

extern "C" void kernel_launch(void* const* d_in, const int* in_sizes, int n_in,
                              void* d_out, int out_size, void* d_ws, size_t ws_size,
                              hipStream_t stream) {
    (void)in_sizes; (void)n_in; (void)out_size; (void)ws_size;

    // setup_inputs() order: boxes f32[B,N,4], images f32[B,H,W,3],
    //                       patch f32[PH,PW,3], transform_decisions i32[B,N,3]
    const float* boxes  = (const float*)d_in[0];
    const float* images = (const float*)d_in[1];
    const float* patch  = (const float*)d_in[2];
    const int*   dec    = (const int*)d_in[3];

    float* out_img = (float*)d_out;                                   // [B,H,W,3]
    float* out_pb  = out_img + (size_t)BATCH * IMH * IMW * 3;         // [B,N,4]
    float* out_dec = out_pb + (size_t)BATCH * NBOX * 4;               // [B,N,3]
    int*   ws_box  = (int*)d_ws;                                      // [B,N,8] = 8 KB

    patch_setup_kernel<<<1, 256, 0, stream>>>(boxes, dec, out_pb, out_dec, ws_box);

    const int blocks = BATCH * (IMH * IMW / 256);                     // 25600
    patch_paste_kernel<<<blocks, 256, 0, stream>>>(images, patch, ws_box, out_img);
}